// DeeperGCN_68521908240970
// MI455X (gfx1250) — compile-verified
//
#include <hip/hip_runtime.h>
#include <hip/hip_bf16.h>
#include <math.h>

#define N_NODES 50000
#define N_EDGES 800000
#define N_GRAPHS 64
#define HIDDEN 128
#define EPS_GEN 1e-7f
#define BN_EPS 1e-5f

typedef float v2f __attribute__((ext_vector_type(2)));
typedef float v8f __attribute__((ext_vector_type(8)));

static inline int nblk(long long total, int bs) { return (int)((total + bs - 1) / bs); }

// ---------------- input encoder: h = x @ W(6x128) + b ----------------
__global__ void k_in_enc(const float* __restrict__ x, const float* __restrict__ w,
                         const float* __restrict__ b, float* __restrict__ h) {
    int idx = blockIdx.x * blockDim.x + threadIdx.x;
    if (idx >= N_NODES * HIDDEN) return;
    int n = idx >> 7, c = idx & 127;
    const float* xr = x + (size_t)n * 6;
    float acc = b[c];
#pragma unroll
    for (int k = 0; k < 6; ++k) acc += xr[k] * w[k * HIDDEN + c];
    h[idx] = acc;
}

// ---------------- batchnorm stats (per-channel sum / sumsq) ----------------
__global__ void k_bn_stats(const float* __restrict__ X, int N, int C,
                           float* __restrict__ sum, float* __restrict__ sumsq) {
    int tid = blockIdx.x * blockDim.x + threadIdx.x;
    int T = gridDim.x * blockDim.x;       // multiple of C (C divides 256)
    int c = tid % C;
    int r = tid / C;
    int rs = T / C;
    float s = 0.f, s2 = 0.f;
    for (; r < N; r += rs) {
        float v = X[(size_t)r * C + c];
        s += v; s2 += v * v;
    }
    atomicAdd(&sum[c], s);
    atomicAdd(&sumsq[c], s2);
}

__global__ void k_bn_fin(const float* __restrict__ sum, const float* __restrict__ sumsq,
                         const float* __restrict__ g, const float* __restrict__ b,
                         int N, int C, float* __restrict__ scale, float* __restrict__ shift) {
    int c = blockIdx.x * blockDim.x + threadIdx.x;
    if (c >= C) return;
    float m = sum[c] / (float)N;
    float v = sumsq[c] / (float)N - m * m;
    float sc = g[c] * rsqrtf(v + BN_EPS);
    scale[c] = sc;
    shift[c] = b[c] - m * sc;
}

// act: 0 = leaky_relu(0.01), 1 = tanh, 2 = relu
__global__ void k_bn_act(const float* __restrict__ X, float* __restrict__ Y,
                         const float* __restrict__ scale, const float* __restrict__ shift,
                         int total, int C, int act) {
    int idx = blockIdx.x * blockDim.x + threadIdx.x;
    if (idx >= total) return;
    int c = idx % C;
    float v = X[idx] * scale[c] + shift[c];
    if (act == 0)      v = v > 0.f ? v : 0.01f * v;
    else if (act == 1) v = tanhf(v);
    else               v = fmaxf(v, 0.f);
    Y[idx] = v;
}

// ---------------- edge phase: msg = relu(z[src])+eps, segment softmax over dst ----------------
__global__ void k_edge_max(const float* __restrict__ Z, const int* __restrict__ src,
                           const int* __restrict__ dst, float* __restrict__ MX,
                           int C, int c4s) {
    long long idx = (long long)blockIdx.x * blockDim.x + threadIdx.x;
    long long total = (long long)N_EDGES << c4s;
    if (idx >= total) return;
    int e  = (int)(idx >> c4s);
    int c4 = (int)(idx & ((1 << c4s) - 1));
    int s = src[e], d = dst[e];
    float4 v = *(const float4*)(Z + (size_t)s * C + c4 * 4);
    int* mp = (int*)(MX + (size_t)d * C + c4 * 4);
    // msg > 0 always => int compare of float bits is order-preserving, init is 0
    atomicMax(mp + 0, __float_as_int(fmaxf(v.x, 0.f) + EPS_GEN));
    atomicMax(mp + 1, __float_as_int(fmaxf(v.y, 0.f) + EPS_GEN));
    atomicMax(mp + 2, __float_as_int(fmaxf(v.z, 0.f) + EPS_GEN));
    atomicMax(mp + 3, __float_as_int(fmaxf(v.w, 0.f) + EPS_GEN));
}

__global__ void k_edge_acc(const float* __restrict__ Z, const int* __restrict__ src,
                           const int* __restrict__ dst, const float* __restrict__ MX,
                           float* __restrict__ SE, float* __restrict__ SEM,
                           int C, int c4s) {
    long long idx = (long long)blockIdx.x * blockDim.x + threadIdx.x;
    long long total = (long long)N_EDGES << c4s;
    if (idx >= total) return;
    int e  = (int)(idx >> c4s);
    int c4 = (int)(idx & ((1 << c4s) - 1));
    int s = src[e], d = dst[e];
    float4 v  = *(const float4*)(Z  + (size_t)s * C + c4 * 4);
    float4 mx = *(const float4*)(MX + (size_t)d * C + c4 * 4);
    size_t o = (size_t)d * C + c4 * 4;
    float m, ev;
    m = fmaxf(v.x, 0.f) + EPS_GEN; ev = expf(m - mx.x); atomicAdd(SE + o + 0, ev); atomicAdd(SEM + o + 0, ev * m);
    m = fmaxf(v.y, 0.f) + EPS_GEN; ev = expf(m - mx.y); atomicAdd(SE + o + 1, ev); atomicAdd(SEM + o + 1, ev * m);
    m = fmaxf(v.z, 0.f) + EPS_GEN; ev = expf(m - mx.z); atomicAdd(SE + o + 2, ev); atomicAdd(SEM + o + 2, ev * m);
    m = fmaxf(v.w, 0.f) + EPS_GEN; ev = expf(m - mx.w); atomicAdd(SE + o + 3, ev); atomicAdd(SEM + o + 3, ev * m);
}

// h2 = agg + z = SEM/(SE+1e-16) + z
__global__ void k_combine(const float* __restrict__ SE, const float* __restrict__ SEM,
                          const float* __restrict__ Z, float* __restrict__ H2, int total) {
    int idx = blockIdx.x * blockDim.x + threadIdx.x;
    if (idx >= total) return;
    H2[idx] = SEM[idx] / (SE[idx] + 1e-16f) + Z[idx];
}

__global__ void k_add(float* __restrict__ A, const float* __restrict__ B, int total) {
    int idx = blockIdx.x * blockDim.x + threadIdx.x;
    if (idx >= total) return;
    A[idx] += B[idx];
}

// ---------------- fp32 WMMA GEMM: Y[N x M] = X[N x K] @ W[K x M] + bias ----------------
// N multiple of 16, K multiple of 4, M multiple of 16. One wave per 16-row block.
__global__ void k_gemm(const float* __restrict__ X, const float* __restrict__ W,
                       const float* __restrict__ B, float* __restrict__ Y,
                       int Nrows, int K, int M) {
    int wave = threadIdx.x >> 5;
    int lane = threadIdx.x & 31;
    int rb = blockIdx.x * (blockDim.x >> 5) + wave;
    int row0 = rb * 16;
    if (row0 >= Nrows) return;          // wave-uniform: EXEC stays all-ones for WMMA
    int half = lane >> 4;               // 0: K pair {k0,k0+1}, 1: K pair {k0+2,k0+3}
    int l15  = lane & 15;
    const float* xrow = X + (size_t)(row0 + l15) * K;

    for (int j = 0; j < M; j += 16) {
        float bias = B[j + l15];        // D col = j + (lane&15) for all 8 acc regs
        v8f acc;
#pragma unroll
        for (int r = 0; r < 8; ++r) acc[r] = bias;

        for (int k0 = 0; k0 < K; k0 += 4) {
            int ka = k0 + half * 2;
            v2f a;                       // A 16x4: lanes 0-15 M rows, half selects K pair
            a.x = xrow[ka];
            a.y = xrow[ka + 1];
            v2f b;                       // B 4x16: lanes = N cols, half selects K pair
            b.x = W[(size_t)ka * M + j + l15];
            b.y = W[(size_t)(ka + 1) * M + j + l15];
            acc = __builtin_amdgcn_wmma_f32_16x16x4_f32(
                false, a, false, b, (short)0, acc, false, false);
        }
        // C/D layout: vgpr r -> row = row0 + half*8 + r, col = j + l15
#pragma unroll
        for (int r = 0; r < 8; ++r) {
            Y[(size_t)(row0 + half * 8 + r) * M + j + l15] = acc[r];
        }
    }
}

// ---------------- pooling + final linear ----------------
__global__ void k_pool(const float* __restrict__ H, const int* __restrict__ batch,
                       float* __restrict__ psum, float* __restrict__ pcnt) {
    int idx = blockIdx.x * blockDim.x + threadIdx.x;
    if (idx >= N_NODES * 16) return;
    int n = idx >> 4, c = idx & 15;
    int g = batch[n];
    atomicAdd(&psum[g * 16 + c], H[idx]);
    if (c == 0) atomicAdd(&pcnt[g], 1.0f);
}

__global__ void k_final(const float* __restrict__ psum, const float* __restrict__ pcnt,
                        const float* __restrict__ lw, const float* __restrict__ lb,
                        float* __restrict__ out) {
    int idx = threadIdx.x;
    if (idx >= N_GRAPHS * 2) return;
    int g = idx >> 1, o = idx & 1;
    float cnt = fmaxf(pcnt[g], 1.0f);
    float acc = lb[o];
#pragma unroll
    for (int c = 0; c < 16; ++c) acc += (psum[g * 16 + c] / cnt) * lw[c * 2 + o];
    out[idx] = acc;
}

// ---------------- host ----------------
extern "C" void kernel_launch(void* const* d_in, const int* in_sizes, int n_in,
                              void* d_out, int out_size, void* d_ws, size_t ws_size,
                              hipStream_t stream) {
    const int N = N_NODES;
    // Identify big arrays by unique size; remaining (in order) are param leaves.
    const float* x = nullptr; const int* ei = nullptr; const int* batch = nullptr;
    const float* P[64]; int psz[64]; int np = 0;
    for (int i = 0; i < n_in; ++i) {
        int s = in_sizes[i];
        if (s == N_NODES * 6 && !x)            x = (const float*)d_in[i];
        else if (s == 2 * N_EDGES && !ei)      ei = (const int*)d_in[i];
        else if (s == N_NODES && !batch)       batch = (const int*)d_in[i];
        else if (np < 64)                      { P[np] = (const float*)d_in[i]; psz[np] = s; ++np; }
    }
    const int* src = ei;
    const int* dst = ei + N_EDGES;

    struct LP { const float *bn_g,*bn_b,*m1w,*m1b,*bn1g,*bn1b,*m2w,*m2b,*bn2g,*bn2b,*m3w,*m3b,*ew,*eb; } L[3];
    const float *ienc_w, *ienc_b, *lin_w, *lin_b;
    bool insertion = (np > 0 && psz[0] == 6 * HIDDEN);   // first leaf is in_enc.w(768) -> insertion order
    if (insertion) {
        // params: in_enc{w,b}, lin{w,b}, layers[ bn_g,bn_b,m1{w,b},bn1{g,b},m2{w,b},bn2{g,b},m3{w,b},enc{w,b} ]
        ienc_w = P[0]; ienc_b = P[1]; lin_w = P[2]; lin_b = P[3];
        for (int l = 0; l < 3; ++l) {
            const float* const* q = P + 4 + 14 * l;
            L[l].bn_g = q[0];  L[l].bn_b = q[1];
            L[l].m1w  = q[2];  L[l].m1b  = q[3];  L[l].bn1g = q[4];  L[l].bn1b = q[5];
            L[l].m2w  = q[6];  L[l].m2b  = q[7];  L[l].bn2g = q[8];  L[l].bn2b = q[9];
            L[l].m3w  = q[10]; L[l].m3b  = q[11]; L[l].ew   = q[12]; L[l].eb   = q[13];
        }
    } else {
        // jax tree order (sorted dict keys): in_enc.b, in_enc.w,
        // layers[ bn_b,bn_g, enc.b,enc.w, m1.b,m1.w, m2.b,m2.w, m3.b,m3.w, bn1b,bn1g, bn2b,bn2g ], lin.b, lin.w
        ienc_b = P[0]; ienc_w = P[1];
        for (int l = 0; l < 3; ++l) {
            const float* const* q = P + 2 + 14 * l;
            L[l].bn_b = q[0];  L[l].bn_g = q[1];  L[l].eb   = q[2];  L[l].ew   = q[3];
            L[l].m1b  = q[4];  L[l].m1w  = q[5];  L[l].m2b  = q[6];  L[l].m2w  = q[7];
            L[l].m3b  = q[8];  L[l].m3w  = q[9];  L[l].bn1b = q[10]; L[l].bn1g = q[11];
            L[l].bn2b = q[12]; L[l].bn2g = q[13];
        }
        lin_b = P[44]; lin_w = P[45];
    }

    // Workspace layout (all fits easily in 192MB L2 per-tensor; total ~205MB)
    char* wsb = (char*)d_ws;
    size_t off = 0;
    auto WS = [&](size_t bytes) -> float* {
        float* p = (float*)(wsb + off);
        off = (off + bytes + 255) & ~(size_t)255;
        return p;
    };
    const size_t NB = (size_t)N * 128 * sizeof(float);
    float* H     = WS(NB);        // h (even layers)
    float* HB    = WS(NB);        // h (odd layers)
    float* Z     = WS(NB);        // act(bn(h)); later reused for z2 (mlp3 out)
    float* MX    = WS(NB);        // segment max; later reused as h2 = agg+z
    float* SESEM = WS(2 * NB);    // SE | SEM; later reused as t2 (N x 2C)
    float* T1    = WS(2 * NB);    // t1 (N x 2C)
    float* s_sum = WS(256 * 4);  float* s_sq    = WS(256 * 4);
    float* s_scl = WS(256 * 4);  float* s_shf   = WS(256 * 4);
    float* psum  = WS(N_GRAPHS * 16 * 4);
    float* pcnt  = WS(N_GRAPHS * 4);
    (void)ws_size; (void)out_size;

    const int gemmGrid = (N / 16 + 7) / 8;   // 8 waves (16-row blocks) per 256-thread block

    // input encoder
    k_in_enc<<<nblk((long long)N * HIDDEN, 256), 256, 0, stream>>>(x, ienc_w, ienc_b, H);

    float* cur = H; float* nxt = HB;
    for (int l = 0; l < 3; ++l) {
        const int C = HIDDEN >> l, C2 = 2 * C, Co = C >> 1;
        const LP lp = L[l];
        const int c4s = (C == 128) ? 5 : (C == 64 ? 4 : 3);   // log2(C/4)
        const long long totE = (long long)N_EDGES << c4s;

        // z = act(bn(h))
        hipMemsetAsync(s_sum, 0, C * 4, stream);
        hipMemsetAsync(s_sq,  0, C * 4, stream);
        k_bn_stats<<<256, 256, 0, stream>>>(cur, N, C, s_sum, s_sq);
        k_bn_fin<<<1, 256, 0, stream>>>(s_sum, s_sq, lp.bn_g, lp.bn_b, N, C, s_scl, s_shf);
        k_bn_act<<<nblk((long long)N * C, 256), 256, 0, stream>>>(cur, Z, s_scl, s_shf, N * C, C, (l == 0) ? 0 : 1);

        // per-channel segment softmax aggregation over dst
        hipMemsetAsync(MX, 0, (size_t)N * C * 4, stream);
        k_edge_max<<<nblk(totE, 256), 256, 0, stream>>>(Z, src, dst, MX, C, c4s);
        hipMemsetAsync(SESEM, 0, (size_t)N * C2 * 4, stream);
        float* SE = SESEM; float* SEM = SESEM + (size_t)N * C;
        k_edge_acc<<<nblk(totE, 256), 256, 0, stream>>>(Z, src, dst, MX, SE, SEM, C, c4s);
        k_combine<<<nblk((long long)N * C, 256), 256, 0, stream>>>(SE, SEM, Z, MX, N * C);  // MX := h2

        // MLP: t1 = relu(bn(h2@W1+b1)); t2 = relu(bn(t1@W2+b2)); z2 = t2@W3+b3
        k_gemm<<<gemmGrid, 256, 0, stream>>>(MX, lp.m1w, lp.m1b, T1, N, C, C2);
        hipMemsetAsync(s_sum, 0, C2 * 4, stream);
        hipMemsetAsync(s_sq,  0, C2 * 4, stream);
        k_bn_stats<<<256, 256, 0, stream>>>(T1, N, C2, s_sum, s_sq);
        k_bn_fin<<<1, 256, 0, stream>>>(s_sum, s_sq, lp.bn1g, lp.bn1b, N, C2, s_scl, s_shf);
        k_bn_act<<<nblk((long long)N * C2, 256), 256, 0, stream>>>(T1, T1, s_scl, s_shf, N * C2, C2, 2);

        k_gemm<<<gemmGrid, 256, 0, stream>>>(T1, lp.m2w, lp.m2b, SESEM, N, C2, C2);        // t2 := SESEM
        hipMemsetAsync(s_sum, 0, C2 * 4, stream);
        hipMemsetAsync(s_sq,  0, C2 * 4, stream);
        k_bn_stats<<<256, 256, 0, stream>>>(SESEM, N, C2, s_sum, s_sq);
        k_bn_fin<<<1, 256, 0, stream>>>(s_sum, s_sq, lp.bn2g, lp.bn2b, N, C2, s_scl, s_shf);
        k_bn_act<<<nblk((long long)N * C2, 256), 256, 0, stream>>>(SESEM, SESEM, s_scl, s_shf, N * C2, C2, 2);

        k_gemm<<<gemmGrid, 256, 0, stream>>>(SESEM, lp.m3w, lp.m3b, Z, N, C2, C);           // z2 := Z

        // h = h + z2 ; h = h @ enc + enc.b (channel halving)
        k_add<<<nblk((long long)N * C, 256), 256, 0, stream>>>(cur, Z, N * C);
        k_gemm<<<gemmGrid, 256, 0, stream>>>(cur, lp.ew, lp.eb, nxt, N, C, Co);
        float* t = cur; cur = nxt; nxt = t;
    }

    // global mean pool + final linear (16 -> 2)
    hipMemsetAsync(psum, 0, N_GRAPHS * 16 * 4, stream);
    hipMemsetAsync(pcnt, 0, N_GRAPHS * 4, stream);
    k_pool<<<nblk((long long)N * 16, 256), 256, 0, stream>>>(cur, batch, psum, pcnt);
    k_final<<<1, 128, 0, stream>>>(psum, pcnt, lin_w, lin_b, (float*)d_out);
}